// TopKRouter_33260226740463
// MI455X (gfx1250) — compile-verified
//
#include <hip/hip_runtime.h>
#include <math.h>

// MI455X (gfx1250) top-2 MoE router.
// Bandwidth-bound: 256MB of x streamed once -> ~11.5us floor @ 23.3 TB/s.
// GEMM [N,4096]x[4096,8] done with V_WMMA_F32_16X16X4_F32 (fp32 precision
// preserved so top-k ordering matches the f32 reference).

typedef __attribute__((ext_vector_type(2))) float v2f;
typedef __attribute__((ext_vector_type(4))) float v4f;
typedef __attribute__((ext_vector_type(8))) float v8f;

#define HDIM 4096
#define NEXP 8
#define TOKENS_PER_WAVE 16
#define WAVES_PER_BLOCK 8
#define TOKENS_PER_BLOCK (TOKENS_PER_WAVE * WAVES_PER_BLOCK)

__global__ __launch_bounds__(256) void router_wmma_kernel(
    const float* __restrict__ x,    // [N, 4096]
    const float* __restrict__ W,    // [8, 4096]
    float* __restrict__ out_rw,     // [N, 2]  routing weights
    float* __restrict__ out_sel,    // [N, 2]  selected experts (as float)
    float* __restrict__ out_rl)     // [N, 8]  router logits
{
    // Per-wave scratch to transpose the D fragment: sc[wave][token][expert]
    __shared__ float sc[WAVES_PER_BLOCK][TOKENS_PER_WAVE][NEXP];

    const int tid  = threadIdx.x;
    const int wave = tid >> 5;
    const int lane = tid & 31;
    const int half = lane >> 4;   // 0 -> K pair {0,1}, 1 -> K pair {2,3}
    const int row  = lane & 15;   // token row for A, expert column for B/D

    const int tokenBase = (blockIdx.x * WAVES_PER_BLOCK + wave) * TOKENS_PER_WAVE;

    // A fragment source: x[tokenBase+row][k + 2*half .. +1]
    const float* aptr = x + (size_t)(tokenBase + row) * HDIM + half * 2;
    // B fragment source: W[row&7][k + 2*half .. +1]; columns 8..15 duplicate
    // experts 0..7 (their D columns are never read).
    const float* bptr = W + (size_t)(row & 7) * HDIM + half * 2;

    v8f acc = {0.f, 0.f, 0.f, 0.f, 0.f, 0.f, 0.f, 0.f};

    // 1024 chained f32 WMMAs; unroll 8 so each token row consumes two full
    // 64B cachelines per unrolled body (good MLP for the HBM stream).
#pragma unroll 8
    for (int k = 0; k < HDIM; k += 4) {
        v2f a = *(const v2f*)(aptr + k);
        v2f b = *(const v2f*)(bptr + k);
        // (neg_a, A, neg_b, B, c_mod, C, reuse_a, reuse_b)
        acc = __builtin_amdgcn_wmma_f32_16x16x4_f32(
            false, a, false, b, (short)0, acc, false, false);
    }

    // D layout: lane in {0..7,16..23} holds expert column (lane&15),
    // VGPR m = token row m (lanes<16) or 8+m (lanes>=16).
    if (row < NEXP) {
#pragma unroll
        for (int m = 0; m < 8; ++m)
            sc[wave][half * 8 + m][row] = acc[m];
    }
    __syncthreads();

    if (lane < TOKENS_PER_WAVE) {
        const int token = tokenBase + lane;
        v4f lo = *(const v4f*)&sc[wave][lane][0];
        v4f hi = *(const v4f*)&sc[wave][lane][4];
        float l[8] = {lo.x, lo.y, lo.z, lo.w, hi.x, hi.y, hi.z, hi.w};

        // top-1 (strict '>' keeps the lowest index on ties, matching
        // jax.lax.top_k ordering)
        int i0 = 0; float b0 = l[0];
#pragma unroll
        for (int e = 1; e < NEXP; ++e)
            if (l[e] > b0) { b0 = l[e]; i0 = e; }
        // top-2
        int i1 = (i0 == 0) ? 1 : 0; float b1 = l[i1];
#pragma unroll
        for (int e = 0; e < NEXP; ++e)
            if (e != i0 && l[e] > b1) { b1 = l[e]; i1 = e; }

        // softmax over the gathered logits [b0, b1]; b0 >= b1 -> stable
        float e1  = expf(b1 - b0);
        float inv = 1.0f / (1.0f + e1);

        *(v2f*)(out_rw  + (size_t)token * 2) = (v2f){inv, e1 * inv};
        *(v2f*)(out_sel + (size_t)token * 2) = (v2f){(float)i0, (float)i1};
        *(v4f*)(out_rl  + (size_t)token * 8)     = lo;
        *(v4f*)(out_rl  + (size_t)token * 8 + 4) = hi;
    }
}

extern "C" void kernel_launch(void* const* d_in, const int* in_sizes, int n_in,
                              void* d_out, int out_size, void* d_ws, size_t ws_size,
                              hipStream_t stream) {
    const float* x = (const float*)d_in[0];   // hidden_states [4,4096,4096] f32
    const float* W = (const float*)d_in[1];   // router weight [8,4096] f32

    const int N = in_sizes[0] / HDIM;         // 16384 tokens

    float* out      = (float*)d_out;
    float* out_rw   = out;                    // [N,2]
    float* out_sel  = out + (size_t)N * 2;    // [N,2]
    float* out_rl   = out + (size_t)N * 4;    // [N,8]

    const int blocks = (N + TOKENS_PER_BLOCK - 1) / TOKENS_PER_BLOCK;  // 128
    router_wmma_kernel<<<dim3(blocks), dim3(256), 0, stream>>>(
        x, W, out_rw, out_sel, out_rl);
}